// SIE_72653666779214
// MI455X (gfx1250) — compile-verified
//
#include <hip/hip_runtime.h>
#include <math.h>

// ---------------------------------------------------------------------------
// x [1,4,2048,2] -> N = 8192 tokens, IN_FEAT=2, HID=4, D4=8.
// Dominant cost: sigmoid-attention (no softmax) fused flash-style with
// V_WMMA_F32_16X16X4_F32.  K/V (256KB each) are L2-resident (192MB L2).
// ---------------------------------------------------------------------------

typedef float v2f __attribute__((ext_vector_type(2)));
typedef float v8f __attribute__((ext_vector_type(8)));

#define N_TOK 8192
#define SEQ   2048
#define TH_VAL 0.5f
#define NKT   (N_TOK / 16)

// Same-wave LDS ops execute in order (ISA: DScnt "in order"), so a
// store->load RAW through LDS needs only compile-time ordering; the
// load->VGPR dependency gets a minimal s_wait_dscnt from the compiler.
#if __has_builtin(__builtin_amdgcn_sched_barrier)
#define LDS_ORDER() __builtin_amdgcn_sched_barrier(0)
#else
#define LDS_ORDER() asm volatile("s_wait_dscnt 0x0" ::: "memory")
#endif

// Fast sigmoid: v_exp_f32 + v_add_f32 + v_rcp_f32.
__device__ __forceinline__ float sigf(float x) {
    return __builtin_amdgcn_rcpf(1.0f + __expf(-x));
}

#if __has_builtin(__builtin_amdgcn_tanhf)
__device__ __forceinline__ float fast_tanh(float x) {
    return __builtin_amdgcn_tanhf(x);           // gfx1250 v_tanh_f32
}
#else
__device__ __forceinline__ float fast_tanh(float x) {
    // tanh(x) = 2*sigmoid(2x) - 1, via hardware exp + rcp
    return 2.0f * __builtin_amdgcn_rcpf(1.0f + __expf(-2.0f * x)) - 1.0f;
}
#endif

__device__ __forceinline__ v8f wmma_f32_16x16x4(v2f a, v2f b, v8f c) {
    // D = A(16x4,f32) * B(4x16,f32) + C(16x16,f32)
    return __builtin_amdgcn_wmma_f32_16x16x4_f32(
        /*neg_a=*/false, a, /*neg_b=*/false, b,
        /*c_mod=*/(short)0, c, /*reuse_a=*/false, /*reuse_b=*/false);
}

// ---------------------------------------------------------------------------
// Kernel 1: positional encoding + input-side projections.
//   gx[n][16] = (x+pe) @ w_ih^T + b_ih + b_hh   (LSTM pre-gates)
//   K [n][8]  = (x+pe) @ fc2_w^T + fc2_b
//   V [n][8]  = (x+pe) @ fc3_w^T + fc3_b
// ---------------------------------------------------------------------------
__global__ void prep_kernel(const float* __restrict__ x,
                            const float* __restrict__ w_ih,
                            const float* __restrict__ b_ih,
                            const float* __restrict__ b_hh,
                            const float* __restrict__ fc2_w,
                            const float* __restrict__ fc2_b,
                            const float* __restrict__ fc3_w,
                            const float* __restrict__ fc3_b,
                            float* __restrict__ gx,
                            float* __restrict__ Km,
                            float* __restrict__ Vm)
{
    int n = blockIdx.x * blockDim.x + threadIdx.x;
    if (n >= N_TOK) return;
    int s = n & (SEQ - 1);
    // feat=2: k=0 -> sin(pos / 10000^0), k=1 -> cos(pos / 10000^1)
    float x0 = x[n * 2 + 0] + sinf((float)s);
    float x1 = x[n * 2 + 1] + cosf((float)s * 1.0e-4f);
#pragma unroll
    for (int j = 0; j < 16; ++j)
        gx[n * 16 + j] = b_ih[j] + b_hh[j] + w_ih[j * 2] * x0 + w_ih[j * 2 + 1] * x1;
#pragma unroll
    for (int j = 0; j < 8; ++j) {
        Km[n * 8 + j] = fc2_b[j] + fc2_w[j * 2] * x0 + fc2_w[j * 2 + 1] * x1;
        Vm[n * 8 + j] = fc3_b[j] + fc3_w[j * 2] * x0 + fc3_w[j * 2 + 1] * x1;
    }
}

// ---------------------------------------------------------------------------
// Kernel 2: inherently-serial 8192-step LSTM scan on ONE wave32.
// Lane j (mod 16) owns gate j; lanes exchange h via wave shuffles.
// Emits Q[t][0..7] = h_t @ fc_w^T + fc_b directly (X never hits memory).
// ---------------------------------------------------------------------------
__global__ void lstm_kernel(const float* __restrict__ gx,
                            const float* __restrict__ w_hh,
                            const float* __restrict__ fc_w,
                            const float* __restrict__ fc_b,
                            float* __restrict__ Qm)
{
    int lane = threadIdx.x;        // 0..31 (wave32)
    int j  = lane & 15;            // gate row (lanes 16..31 replicate 0..15)
    int jq = lane & 7;             // Q row
    int k  = lane & 3;             // hidden index this lane maintains

    float w0 = w_hh[j * 4 + 0], w1 = w_hh[j * 4 + 1];
    float w2 = w_hh[j * 4 + 2], w3 = w_hh[j * 4 + 3];
    float fw0 = fc_w[jq * 4 + 0], fw1 = fc_w[jq * 4 + 1];
    float fw2 = fc_w[jq * 4 + 2], fw3 = fc_w[jq * 4 + 3];
    float fb = fc_b[jq];

    float h = 0.0f, c = 0.0f;
    float gcur = gx[j];
    for (int t = 0; t < N_TOK; ++t) {
        // prefetch next pre-gate off the critical recurrence path
        float gnext = (t + 1 < N_TOK) ? gx[(t + 1) * 16 + j] : 0.0f;

        float h0 = __shfl(h, 0, 32), h1 = __shfl(h, 1, 32);
        float h2 = __shfl(h, 2, 32), h3 = __shfl(h, 3, 32);
        float g = gcur + w0 * h0 + w1 * h1 + w2 * h2 + w3 * h3;
        float a = (j >= 8 && j < 12) ? fast_tanh(g) : sigf(g); // gates i,f,g,o

        float ig = __shfl(a, k, 32);
        float fg = __shfl(a, k + 4, 32);
        float gg = __shfl(a, k + 8, 32);
        float og = __shfl(a, k + 12, 32);
        c = fg * c + ig * gg;
        h = og * fast_tanh(c);

        float hh0 = __shfl(h, 0, 32), hh1 = __shfl(h, 1, 32);
        float hh2 = __shfl(h, 2, 32), hh3 = __shfl(h, 3, 32);
        if (lane < 8)
            Qm[t * 8 + lane] = fb + fw0 * hh0 + fw1 * hh1 + fw2 * hh2 + fw3 * hh3;

        gcur = gnext;
    }
}

// ---------------------------------------------------------------------------
// Kernel 3: fused flash-style sigmoid attention + threshold + output proj.
//   out[i] = relu_th( sum_j sigmoid(q_i . k_j) * v_j ) @ fcout_w^T + fcout_b
// One wave per 16-row Q tile; loop over 512 KV tiles.
// Per tile-pair: 2x WMMA (Q.K^T, Kdim=8) + 4x WMMA (S.Vpad, Kdim=16).
// K/V tile loads are software-pipelined one iteration ahead; the S tile
// D->A relayout goes through a wave-private LDS slice ordered only at
// compile time (hardware keeps same-wave DS ops in order).
// ---------------------------------------------------------------------------
__global__ __launch_bounds__(128) void attn_kernel(
        const float* __restrict__ Qm,
        const float* __restrict__ Km,
        const float* __restrict__ Vm,
        const float* __restrict__ fcout_w,
        const float* __restrict__ fcout_b,
        float* __restrict__ out)
{
    __shared__ float lds[4][256];              // 1KB slice per wave
    const int wave = threadIdx.x >> 5;
    const int lane = threadIdx.x & 31;
    float* slice = &lds[wave][0];

    const int qtile = blockIdx.x * 4 + wave;   // 512 tiles total
    const int q0 = qtile * 16;
    const int r  = lane & 15;                  // M (or N) index within tile
    const int hi = lane >> 4;                  // lane half selects K pair
    const int ko = hi * 2;
    const int rc = r & 7;                      // clamped feature col
    const bool rlo = (r < 8);

    // A operands for the Q tile: 16x8 -> two 16x4 chunks.
    // ISA 16x4 f32 A-layout: lane holds row r, VGPR v holds K = ko + v.
    v2f qa0 = *(const v2f*)(Qm + (q0 + r) * 8 + ko);
    v2f qa1 = *(const v2f*)(Qm + (q0 + r) * 8 + 4 + ko);

    v8f acc = {0.f, 0.f, 0.f, 0.f, 0.f, 0.f, 0.f, 0.f};

    // ---- software pipeline: preload K/V operands for tile 0 ----
    // K^T B-operands: B[f][n] = K[n][f]; lane holds col r, K pair at ko.
    v2f kb0 = *(const v2f*)(Km + r * 8 + ko);
    v2f kb1 = *(const v2f*)(Km + r * 8 + 4 + ko);
    // V B-operand elements for the 4 S@Vpad chunks:
    //   ve[2c+v] = V[4c + ko + v][r]  (zero for padded cols r>=8)
    float ve[8];
#pragma unroll
    for (int cch = 0; cch < 4; ++cch) {
        float e0 = Vm[(cch * 4 + ko + 0) * 8 + rc];
        float e1 = Vm[(cch * 4 + ko + 1) * 8 + rc];
        ve[cch * 2 + 0] = rlo ? e0 : 0.0f;
        ve[cch * 2 + 1] = rlo ? e1 : 0.0f;
    }

    for (int kt = 0; kt < NKT; ++kt) {
        // ---- issue NEXT tile's global loads first (overlap everything) ----
        const int ktn = (kt + 1 < NKT) ? kt + 1 : kt;   // clamp (dead result)
        const int kvn = ktn * 16;
        v2f nk0 = *(const v2f*)(Km + (kvn + r) * 8 + ko);
        v2f nk1 = *(const v2f*)(Km + (kvn + r) * 8 + 4 + ko);
        float nv[8];
#pragma unroll
        for (int cch = 0; cch < 4; ++cch) {
            float e0 = Vm[(kvn + cch * 4 + ko + 0) * 8 + rc];
            float e1 = Vm[(kvn + cch * 4 + ko + 1) * 8 + rc];
            nv[cch * 2 + 0] = rlo ? e0 : 0.0f;
            nv[cch * 2 + 1] = rlo ? e1 : 0.0f;
        }
        if (kt + 8 < NKT) {                    // keep tiles streaming WGP-near
            __builtin_prefetch(Km + (kt * 16 + 128 + r) * 8, 0, 3);
            __builtin_prefetch(Vm + (kt * 16 + 128 + r) * 8, 0, 3);
        }

        // ---- compute with CURRENT tile (kb0/kb1/ve) ----
        v8f s = {0.f, 0.f, 0.f, 0.f, 0.f, 0.f, 0.f, 0.f};
        s = wmma_f32_16x16x4(qa0, kb0, s);
        s = wmma_f32_16x16x4(qa1, kb1, s);

        // elementwise sigmoid on the 16x16 logits tile (exp + rcp only)
#pragma unroll
        for (int i = 0; i < 8; ++i)
            s[i] = sigf(s[i]);

        // Stage S (C/D layout: vgpr i -> row i+8*hi, lane -> col r) into LDS,
        // then reload in A layout for the S @ Vpad accumulation.
#pragma unroll
        for (int i = 0; i < 8; ++i)
            slice[(i + 8 * hi) * 16 + r] = s[i];
        LDS_ORDER();                           // compile-time store->load order

#pragma unroll
        for (int cch = 0; cch < 4; ++cch) {
            v2f sa = *(const v2f*)(slice + r * 16 + cch * 4 + ko);
            v2f vb;
            vb.x = ve[cch * 2 + 0];
            vb.y = ve[cch * 2 + 1];
            acc = wmma_f32_16x16x4(sa, vb, acc);
        }
        LDS_ORDER();                           // reads stay before next stores

        // ---- rotate pipeline registers ----
        kb0 = nk0; kb1 = nk1;
#pragma unroll
        for (int i = 0; i < 8; ++i) ve[i] = nv[i];
    }

    // Epilogue: threshold-ReLU + out(16x8) @ fcout_w^T(8x2) + fcout_b.
#pragma unroll
    for (int i = 0; i < 8; ++i)
        slice[(i + 8 * hi) * 16 + r] = acc[i];
    LDS_ORDER();

    {
        const int m = r;           // output row within tile
        const int o = hi;          // output feature 0/1 (32 lanes = 16x2)
        float res = fcout_b[o];
#pragma unroll
        for (int f = 0; f < 8; ++f) {
            float v = slice[m * 16 + f];
            v = (v > TH_VAL) ? v : 0.0f;
            res += v * fcout_w[o * 8 + f];
        }
        out[(q0 + m) * 2 + o] = res;
    }
}

// ---------------------------------------------------------------------------
extern "C" void kernel_launch(void* const* d_in, const int* in_sizes, int n_in,
                              void* d_out, int out_size, void* d_ws, size_t ws_size,
                              hipStream_t stream)
{
    const float* x       = (const float*)d_in[0];
    const float* w_ih    = (const float*)d_in[1];
    const float* w_hh    = (const float*)d_in[2];
    const float* b_ih    = (const float*)d_in[3];
    const float* b_hh    = (const float*)d_in[4];
    const float* fc_w    = (const float*)d_in[5];
    const float* fc_b    = (const float*)d_in[6];
    const float* fc2_w   = (const float*)d_in[7];
    const float* fc2_b   = (const float*)d_in[8];
    const float* fc3_w   = (const float*)d_in[9];
    const float* fc3_b   = (const float*)d_in[10];
    const float* fcout_w = (const float*)d_in[11];
    const float* fcout_b = (const float*)d_in[12];

    float* ws = (float*)d_ws;
    float* gx = ws;                 // 8192*16 = 131072 floats
    float* Qm = ws + 131072;        // 8192*8
    float* Km = ws + 196608;        // 8192*8
    float* Vm = ws + 262144;        // 8192*8   (total 1.25 MB)

    prep_kernel<<<N_TOK / 256, 256, 0, stream>>>(x, w_ih, b_ih, b_hh,
                                                 fc2_w, fc2_b, fc3_w, fc3_b,
                                                 gx, Km, Vm);
    lstm_kernel<<<1, 32, 0, stream>>>(gx, w_hh, fc_w, fc_b, Qm);
    attn_kernel<<<(N_TOK / 16) / 4, 128, 0, stream>>>(Qm, Km, Vm,
                                                      fcout_w, fcout_b,
                                                      (float*)d_out);
}